// CRFCNN_43396349559059
// MI455X (gfx1250) — compile-verified
//
#include <hip/hip_runtime.h>
#include <hip/hip_bf16.h>

// ---------------------------------------------------------------------------
// Fused CRF-CNN for gfx1250 (MI455X).
// Stage 1: per-(batch,channel) plane, conv5x5+BN fused into z (registers),
//          50 CRF iterations ping-ponged in LDS (zero border = SAME /9 pad),
//          8x2 tiles/thread with separable sums (conflict-free stride-2 lanes),
//          3x3/3 maxpool -> ws.
// Stage 2: per-image, conv2 as WMMA f32 16x16x4 GEMM (M=1369 px, N=16 (10
//          used), K=800) from LDS, then LDS-resident tiled CRF + pool + mean.
// Stage 3: log_softmax over 10 classes.
// ---------------------------------------------------------------------------

typedef __attribute__((ext_vector_type(2))) float v2f;
typedef __attribute__((ext_vector_type(8))) float v8f;

__device__ __forceinline__ float sigm(float t) {
    return 1.0f / (1.0f + __expf(-t));
}

// ------------------------------ Stage 1 ------------------------------------
// grid = 128*32 blocks, block = 1024 threads.
// LDS: x 128*128 (16384) | qA 126*126 (15876) | qB 126*126 (15876) | slack 640
#define S1_Q 15876
#define S1_LDS_FLOATS (16384 + 2 * S1_Q + 640)

__global__ __launch_bounds__(1024, 1)
void crf_stage1(const float* __restrict__ x, const float* __restrict__ w1,
                const float* __restrict__ b1, const float* __restrict__ g1,
                const float* __restrict__ be1, const float* __restrict__ m1,
                const float* __restrict__ v1, float* __restrict__ pool1) {
    extern __shared__ float lds[];
    float* xs = lds;                 // 16384
    float* qA = lds + 16384;         // 15876 (padded 126x126, zero border)
    float* qB = qA + S1_Q;           // 15876 (+640 slack after for patch overread)

    const int bo  = blockIdx.x;      // b*32 + oc
    const int oc  = bo & 31;
    const int tid = threadIdx.x;

    const float* xp = x + (size_t)(bo >> 5) * (128 * 128);
    for (int i = tid; i < 128 * 128; i += 1024) {
        __builtin_prefetch(xp + i + 8192, 0, 1);
        xs[i] = xp[i];
    }
    // Zero q buffers + slack (borders stay 0 forever -> SAME pad, /9).
    for (int i = tid; i < 2 * S1_Q + 640; i += 1024) qA[i] = 0.0f;

    // Fold BN into conv: z = conv(x, w1*sc) + ((b1-m1)*sc + be1)
    const float sc = g1[oc] * rsqrtf(v1[oc] + 1e-5f);
    const float sh = (b1[oc] - m1[oc]) * sc + be1[oc];
    float wf[25];
#pragma unroll
    for (int j = 0; j < 25; ++j) wf[j] = w1[oc * 25 + j] * sc;

    __syncthreads();

    // 8x2 tile per thread: 16 row-bands x 62 col-tiles = 992 tiles.
    const bool active = tid < 992;
    const int  tr = tid / 62;            // 0..15 (band 15 rows 120..127, 124+ invalid)
    const int  tc = tid - tr * 62;       // 0..61
    const int  r0 = 8 * tr;
    const int  c0 = 2 * tc;
    const int  tbase = r0 * 126 + c0;    // padded patch origin

    float z[16];
#pragma unroll
    for (int k = 0; k < 16; ++k) z[k] = 0.0f;

    if (active) {
#pragma unroll
        for (int i = 0; i < 8; ++i) {
            const int r = r0 + i;
            if (r < 124) {
#pragma unroll
                for (int j = 0; j < 2; ++j) {
                    const int c = c0 + j;
                    float acc = sh;
#pragma unroll
                    for (int ky = 0; ky < 5; ++ky)
#pragma unroll
                        for (int kx = 0; kx < 5; ++kx)
                            acc = fmaf(xs[(r + ky) * 128 + (c + kx)],
                                       wf[ky * 5 + kx], acc);
                    z[i * 2 + j] = acc;
                    qA[tbase + (i + 1) * 126 + (j + 1)] = sigm(acc);
                }
            }
        }
    }
    __syncthreads();

    // 50 mean-field iterations: 10x4 patch -> vertical sums -> 16 outputs.
    float* qs = qA;
    float* qd = qB;
    for (int it = 0; it < 50; ++it) {
        if (active) {
            float vs[8][4];
#pragma unroll
            for (int cc = 0; cc < 4; ++cc) {
                float p[10];
#pragma unroll
                for (int k = 0; k < 10; ++k) p[k] = qs[tbase + k * 126 + cc];
#pragma unroll
                for (int i = 0; i < 8; ++i) vs[i][cc] = p[i] + p[i + 1] + p[i + 2];
            }
#pragma unroll
            for (int i = 0; i < 8; ++i) {
                if (r0 + i < 124) {
#pragma unroll
                    for (int j = 0; j < 2; ++j) {
                        const float s = vs[i][j] + vs[i][j + 1] + vs[i][j + 2];
                        qd[tbase + (i + 1) * 126 + (j + 1)] =
                            sigm(fmaf(s, 1.0f / 9.0f, z[i * 2 + j]));
                    }
                }
            }
        }
        float* t = qs; qs = qd; qd = t;
        __syncthreads();
    }

    // 3x3 stride-3 VALID maxpool -> 41x41.
    float* outp = pool1 + (size_t)bo * 1681;
    for (int i = tid; i < 1681; i += 1024) {
        const int orow = i / 41;
        const int ocol = i - orow * 41;
        const int base = (3 * orow + 1) * 126 + (3 * ocol + 1);
        float mx = qs[base];
        mx = fmaxf(mx, qs[base + 1]);
        mx = fmaxf(mx, qs[base + 2]);
        mx = fmaxf(mx, qs[base + 126]);
        mx = fmaxf(mx, qs[base + 127]);
        mx = fmaxf(mx, qs[base + 128]);
        mx = fmaxf(mx, qs[base + 252]);
        mx = fmaxf(mx, qs[base + 253]);
        mx = fmaxf(mx, qs[base + 254]);
        outp[i] = mx;
    }
}

// ------------------------------ Stage 2 ------------------------------------
// grid = 128 blocks (one per image), block = 1024 threads.
// LDS floats: region0 53824 (input planes, later qA/qB 2*10*1680)
//             z 13690 | w2 8800 (last 800 = zero row) | ktab 800 | zs/zh/red
#define S2_REGION0 53824
#define S2_PITCH 40
#define S2_PLANE (42 * S2_PITCH)   // 1680 floats per padded plane
#define S2_QBUF (10 * S2_PLANE)    // 16800
#define S2_LDS_FLOATS (S2_REGION0 + 13690 + 8800 + 800 + 16 + 16 + 320)

__global__ __launch_bounds__(1024, 1)
void crf_stage2(const float* __restrict__ pool1, const float* __restrict__ w2,
                const float* __restrict__ b2, const float* __restrict__ g2,
                const float* __restrict__ be2, const float* __restrict__ m2,
                const float* __restrict__ v2a, float* __restrict__ means) {
    extern __shared__ float lds[];
    float* reg0 = lds;                          // input planes OR qA/qB
    float* zbuf = lds + S2_REGION0;             // 13690 = 10*1369
    float* wbuf = zbuf + 13690;                 // 8800 = 10*800 + zero row
    int*   ktab = (int*)(wbuf + 8800);          // 800
    float* zsc  = (float*)(ktab + 800);         // 16
    float* zsh  = zsc + 16;                     // 16
    float* red  = zsh + 16;                     // 320

    const int b   = blockIdx.x;
    const int tid = threadIdx.x;

    const float* ip = pool1 + (size_t)b * 53824;
    for (int i = tid; i < 53824; i += 1024) {
        __builtin_prefetch(ip + i + 8192, 0, 1);
        reg0[i] = ip[i];
    }
    // Weights + one zeroed row for invalid output channels (n >= 10).
    for (int i = tid; i < 8800; i += 1024) wbuf[i] = (i < 8000) ? w2[i] : 0.0f;
    // im2col offset table: k = c*25 + ky*5 + kx -> c*1681 + ky*41 + kx
    if (tid < 800) {
        const int c  = tid / 25;
        const int rr = tid - c * 25;
        const int ky = rr / 5;
        const int kx = rr - ky * 5;
        ktab[tid] = c * 1681 + ky * 41 + kx;
    }
    if (tid < 16) {
        if (tid < 10) {
            const float s = g2[tid] * rsqrtf(v2a[tid] + 1e-5f);
            zsc[tid] = s;
            zsh[tid] = (b2[tid] - m2[tid]) * s + be2[tid];
        } else {
            zsc[tid] = 0.0f;
            zsh[tid] = 0.0f;
        }
    }
    __syncthreads();

    // --- conv2 as WMMA f32 16x16x4 GEMM: M=1369 px, N=16 (10 used), K=800 ---
    const int lane  = tid & 31;
    const int wv    = tid >> 5;        // wave id 0..31
    const int mA    = lane & 15;       // A: M index; B/D: N index
    const int kl    = (lane >> 4) * 2; // K sub-offset per half-wave
    const int wbase = (mA < 10) ? mA * 800 : 8000;   // zero row for n >= 10

    for (int t = wv; t < 86; t += 32) {        // 86 = ceil(1369/16) M-tiles
        const int p = t * 16 + mA;
        int abase = 0;
        if (p < 1369) {
            const int pr = p / 37;
            const int pc = p - pr * 37;
            abase = pr * 41 + pc;
        }
        v8f acc = {};
        for (int kk = 0; kk < 800; kk += 4) {
            const int k0 = kk + kl;
            v2f A, B;
            A.x = reg0[ktab[k0]     + abase];
            A.y = reg0[ktab[k0 + 1] + abase];
            B.x = wbuf[wbase + k0];
            B.y = wbuf[wbase + k0 + 1];
            acc = __builtin_amdgcn_wmma_f32_16x16x4_f32(
                false, A, false, B, (short)0, acc, false, false);
        }
        // D layout: VGPR v -> M = v (lanes 0-15) or v+8 (lanes 16-31), N = lane&15
#pragma unroll
        for (int vI = 0; vI < 8; ++vI) {
            const int pm = t * 16 + vI + ((lane >> 4) << 3);
            if (mA < 10 && pm < 1369)
                zbuf[mA * 1369 + pm] = acc[vI] * zsc[mA] + zsh[mA];
        }
    }
    __syncthreads();

    // Reuse region0 for padded (42x40) q ping-pong buffers; zero everything.
    float* qA = reg0;            // 16800
    float* qB = reg0 + S2_QBUF;  // 16800
    for (int i = tid; i < 2 * S2_QBUF; i += 1024) reg0[i] = 0.0f;
    __syncthreads();

    // 8x2 tiles: 10 planes x (5 bands x 19 col-tiles) = 950 threads.
    const bool active = tid < 950;
    const int  ocn = tid / 95;
    const int  ti  = tid - ocn * 95;
    const int  tr  = ti / 19;            // 0..4 (band 4 rows 32..39, 37+ invalid)
    const int  tc  = ti - tr * 19;       // 0..18 (tile 18: col 37 invalid)
    const int  r0  = 8 * tr;
    const int  c0  = 2 * tc;
    const int  tbase = ocn * S2_PLANE + r0 * S2_PITCH + c0;

    float zv[16];
#pragma unroll
    for (int k = 0; k < 16; ++k) zv[k] = 0.0f;

    if (active) {
#pragma unroll
        for (int i = 0; i < 8; ++i) {
            const int r = r0 + i;
            if (r < 37) {
#pragma unroll
                for (int j = 0; j < 2; ++j) {
                    const int c = c0 + j;
                    if (c < 37) {
                        const float zz = zbuf[ocn * 1369 + r * 37 + c];
                        zv[i * 2 + j] = zz;
                        qA[tbase + (i + 1) * S2_PITCH + (j + 1)] = sigm(zz);
                    }
                }
            }
        }
    }
    __syncthreads();

    float* qs = qA;
    float* qd = qB;
    for (int it = 0; it < 50; ++it) {
        if (active) {
            float vs[8][4];
#pragma unroll
            for (int cc = 0; cc < 4; ++cc) {
                float p[10];
#pragma unroll
                for (int k = 0; k < 10; ++k) p[k] = qs[tbase + k * S2_PITCH + cc];
#pragma unroll
                for (int i = 0; i < 8; ++i) vs[i][cc] = p[i] + p[i + 1] + p[i + 2];
            }
#pragma unroll
            for (int i = 0; i < 8; ++i) {
                if (r0 + i < 37) {
#pragma unroll
                    for (int j = 0; j < 2; ++j) {
                        if (c0 + j < 37) {
                            const float s = vs[i][j] + vs[i][j + 1] + vs[i][j + 2];
                            qd[tbase + (i + 1) * S2_PITCH + (j + 1)] =
                                sigm(fmaf(s, 1.0f / 9.0f, zv[i * 2 + j]));
                        }
                    }
                }
            }
        }
        float* tm = qs; qs = qd; qd = tm;
        __syncthreads();
    }

    // 2x2 stride-2 maxpool (37->18) + spatial mean, deterministic reduction.
    if (tid < 320) {
        const int oc = tid >> 5;
        const int ln = tid & 31;
        float ssum = 0.0f;
        for (int w = ln; w < 324; w += 32) {
            const int orow = w / 18;
            const int ocol = w - orow * 18;
            const int base = oc * S2_PLANE + (2 * orow + 1) * S2_PITCH + (2 * ocol + 1);
            const float mx = fmaxf(fmaxf(qs[base], qs[base + 1]),
                                   fmaxf(qs[base + S2_PITCH], qs[base + S2_PITCH + 1]));
            ssum += mx;
        }
        red[tid] = ssum;
    }
    __syncthreads();
    if (tid < 10) {
        float s = 0.0f;
        for (int j = 0; j < 32; ++j) s += red[tid * 32 + j];
        means[b * 10 + tid] = s * (1.0f / 324.0f);
    }
}

// ------------------------------ Stage 3 ------------------------------------
__global__ void logsm_kernel(const float* __restrict__ means,
                             float* __restrict__ out) {
    const int r = threadIdx.x;
    if (r < 128) {
        float v[10];
        float mx = -3.4e38f;
#pragma unroll
        for (int j = 0; j < 10; ++j) {
            v[j] = means[r * 10 + j];
            mx = fmaxf(mx, v[j]);
        }
        float s = 0.0f;
#pragma unroll
        for (int j = 0; j < 10; ++j) s += __expf(v[j] - mx);
        const float lse = mx + __logf(s);
#pragma unroll
        for (int j = 0; j < 10; ++j) out[r * 10 + j] = v[j] - lse;
    }
}

// ---------------------------------------------------------------------------
extern "C" void kernel_launch(void* const* d_in, const int* in_sizes, int n_in,
                              void* d_out, int out_size, void* d_ws, size_t ws_size,
                              hipStream_t stream) {
    (void)in_sizes; (void)n_in; (void)out_size; (void)ws_size;

    const float* x   = (const float*)d_in[0];
    const float* w1  = (const float*)d_in[1];
    const float* b1  = (const float*)d_in[2];
    const float* g1  = (const float*)d_in[3];
    const float* be1 = (const float*)d_in[4];
    const float* m1  = (const float*)d_in[5];
    const float* v1  = (const float*)d_in[6];
    const float* w2  = (const float*)d_in[7];
    const float* b2  = (const float*)d_in[8];
    const float* g2  = (const float*)d_in[9];
    const float* be2 = (const float*)d_in[10];
    const float* m2  = (const float*)d_in[11];
    const float* v2a = (const float*)d_in[12];
    float* out = (float*)d_out;

    float* pool1 = (float*)d_ws;                      // 128*32*41*41 floats
    float* means = pool1 + (size_t)128 * 32 * 1681;   // 128*10 floats

    const int s1_bytes = S1_LDS_FLOATS * 4;
    const int s2_bytes = S2_LDS_FLOATS * 4;
    (void)hipFuncSetAttribute((const void*)crf_stage1,
                              hipFuncAttributeMaxDynamicSharedMemorySize, s1_bytes);
    (void)hipFuncSetAttribute((const void*)crf_stage2,
                              hipFuncAttributeMaxDynamicSharedMemorySize, s2_bytes);

    crf_stage1<<<128 * 32, 1024, s1_bytes, stream>>>(x, w1, b1, g1, be1, m1, v1, pool1);
    crf_stage2<<<128, 1024, s2_bytes, stream>>>(pool1, w2, b2, g2, be2, m2, v2a, means);
    logsm_kernel<<<1, 128, 0, stream>>>(means, out);
}